// b_squared_layer_8229157339488
// MI455X (gfx1250) — compile-verified
//
#include <hip/hip_runtime.h>

typedef __attribute__((ext_vector_type(2))) float v2f;
typedef __attribute__((ext_vector_type(8))) float v8f;

// ---------------------------------------------------------------------------
// Kernel 1: channel-sum-of-squares via V_WMMA_F32_16X16X4_F32.
// One wave (32 lanes) per tile of 16 spatial positions. 784 positions = 49
// tiles. For each 4-channel chunk, A[m,k] = x[pos m, chan k]; the fp32 A
// (16x4) and B (4x16) per-lane layouts coincide (M/N = lane%16,
// K = 2*(lane/16)+vgpr), so the same v2f feeds both operands and
// D = A*A^T accumulates sum-of-squares on the diagonal, exactly in fp32.
// ---------------------------------------------------------------------------
__global__ __launch_bounds__(32) void fsq_wmma_kernel(
    const float* __restrict__ x,   // x[0] base: [28*28 positions][512 channels]
    float* __restrict__ fsq)       // [784]
{
    const int lane = threadIdx.x;        // 0..31, full wave active (EXEC all 1s)
    const int p0   = blockIdx.x * 16;    // first position of this tile
    const int m    = lane & 15;          // A-matrix row / B-matrix col
    const int half = lane >> 4;          // 0: K=0,1   1: K=2,3

    const float* base = x + (size_t)(p0 + m) * 512 + 2 * half;

    v8f c = {};
#pragma unroll 8
    for (int k = 0; k < 512; k += 4) {
        v2f a = *(const v2f*)(base + k);  // A[m, k+2h], A[m, k+2h+1]
        // D = A x A^T + C   (fp32 WMMA, round-to-nearest-even)
        c = __builtin_amdgcn_wmma_f32_16x16x4_f32(
                /*neg_a=*/false, a, /*neg_b=*/false, a,
                /*c_mod=*/(short)0, c, /*reuse_a=*/false, /*reuse_b=*/false);
    }

    // Diagonal extraction (C/D layout: lanes 0-15 -> M=vgpr,N=lane;
    // lanes 16-31 -> M=vgpr+8, N=lane-16).
    //   m in 0..7  : lane = m,      vgpr = m
    //   m in 8..15 : lane = m + 16, vgpr = m - 8   (i.e. lanes 24..31)
    const int idx = lane & 7;
    float val = 0.0f;
#pragma unroll
    for (int q = 0; q < 8; ++q)
        if (idx == q) val = c[q];

    if (lane < 8)        fsq[p0 + lane]        = val;
    else if (lane >= 24) fsq[p0 + (lane - 16)] = val;
}

// ---------------------------------------------------------------------------
// Kernel 2: integral image + clipped box sums + batch broadcast.
// Single 1024-thread block; all data fits trivially in LDS.
// ---------------------------------------------------------------------------
__global__ __launch_bounds__(1024) void integral_box_kernel(
    const float* __restrict__ fsq,  // [784] = [28][28]
    float* __restrict__ out)        // [32 * 29 * 29]
{
    __shared__ float I[29 * 29];    // integral image, I[0][*] = I[*][0] = 0
    __shared__ float o2[29 * 29];   // out2d

    const int tid = threadIdx.x;

    // Load fsq into I[r+1][c+1]; zero the top row / left column.
    if (tid < 784) {
        int r = tid / 28, cc = tid % 28;
        I[(r + 1) * 29 + (cc + 1)] = fsq[tid];
    }
    if (tid < 29) { I[tid] = 0.0f; I[tid * 29] = 0.0f; }
    __syncthreads();

    // Row-wise cumulative sum (28 independent serial scans, 28 elems each).
    if (tid < 28) {
        int r = tid + 1;
        float s = 0.0f;
        for (int cc = 1; cc <= 28; ++cc) { s += I[r * 29 + cc]; I[r * 29 + cc] = s; }
    }
    __syncthreads();

    // Column-wise cumulative sum.
    if (tid < 28) {
        int cc = tid + 1;
        float s = 0.0f;
        for (int r = 1; r <= 28; ++r) { s += I[r * 29 + cc]; I[r * 29 + cc] = s; }
    }
    __syncthreads();

    // Clipped box sums: out2d[i,j] = I[r1,c1]-I[r0,c1]-I[r1,c0]+I[r0,c0]
    if (tid < 841) {
        int i = tid / 29, j = tid % 29;
        int r0 = min(max(14 - i, 0), 28);
        int r1 = min(max(42 - i, 0), 28);
        int c0 = min(max(14 - j, 0), 28);
        int c1 = min(max(42 - j, 0), 28);
        o2[tid] = I[r1 * 29 + c1] - I[r0 * 29 + c1]
                - I[r1 * 29 + c0] + I[r0 * 29 + c0];
    }
    __syncthreads();

    // Broadcast across the 32 batch elements (mask is batch-constant).
    for (int idx = tid; idx < 32 * 841; idx += 1024)
        out[idx] = o2[idx % 841];
}

// ---------------------------------------------------------------------------
extern "C" void kernel_launch(void* const* d_in, const int* in_sizes, int n_in,
                              void* d_out, int out_size, void* d_ws, size_t ws_size,
                              hipStream_t stream) {
    (void)in_sizes; (void)n_in; (void)out_size; (void)ws_size;

    const float* x  = (const float*)d_in[0];  // (32,28,28,512) f32; only x[0] used
    float* out      = (float*)d_out;          // (32,29,29,1) f32
    float* fsq_ws   = (float*)d_ws;           // 784 floats of scratch

    fsq_wmma_kernel<<<49, 32, 0, stream>>>(x, fsq_ws);
    integral_box_kernel<<<1, 1024, 0, stream>>>(fsq_ws, out);
}